// GraphEmb_19284403159293
// MI455X (gfx1250) — compile-verified
//
#include <hip/hip_runtime.h>
#include <hip/hip_bf16.h>

#define NDIM    128
#define GDIM    128
#define NGRAPH  16384

typedef __attribute__((ext_vector_type(2))) float v2f;
typedef __attribute__((ext_vector_type(8))) float v8f;

// ---------------------------------------------------------------------------
// Kernel 1: zero the workspace accumulators (sG [16384x128] ++ cG [16384])
// ---------------------------------------------------------------------------
__global__ void zero_ws_kernel(float* __restrict__ ws, int n) {
    int i = blockIdx.x * blockDim.x + threadIdx.x;
    int stride = gridDim.x * blockDim.x;
    for (; i < n; i += stride) ws[i] = 0.0f;
}

// ---------------------------------------------------------------------------
// Kernel 2: streaming gate + weighted segment accumulation.
// One wave (32 lanes) owns a contiguous chunk of nodes; lane l holds features
// [4l..4l+3] of each 128-wide row. Gate dot via wave32 butterfly reduction.
// batch[] is sorted -> accumulate in registers, flush atomics on id change.
// ---------------------------------------------------------------------------
__global__ __launch_bounds__(256) void gate_scatter_kernel(
    const float* __restrict__ h, const int* __restrict__ batch,
    const float* __restrict__ Wg, const float* __restrict__ bg,
    float* __restrict__ sG, float* __restrict__ cG,
    int nNodes, int nodesPerWave)
{
    const int lane = threadIdx.x & 31;
    const int wave = (blockIdx.x * blockDim.x + threadIdx.x) >> 5;

    long start = (long)wave * nodesPerWave;
    long end   = start + nodesPerWave;
    if (end > nNodes) end = nNodes;
    if (start >= end) return;

    const float4 wg4 = *(const float4*)(Wg + lane * 4);
    const float  bgs = bg[0];

    float4 acc  = make_float4(0.f, 0.f, 0.f, 0.f);
    float  gsum = 0.f;
    int    cur  = batch[start];

    for (long v = start; v < end; ++v) {
        const float4 hv = *(const float4*)(h + v * (long)NDIM + lane * 4);

        // gate: sigma(h_v . W_g + b_g); full 128-dim dot across the wave
        float p = hv.x * wg4.x + hv.y * wg4.y + hv.z * wg4.z + hv.w * wg4.w;
        #pragma unroll
        for (int off = 16; off > 0; off >>= 1)
            p += __shfl_xor(p, off, 32);
        const float g = 1.0f / (1.0f + __expf(-(p + bgs)));

        const int gid = batch[v];           // wave-uniform (sorted input)
        if (gid != cur) {
            float* dst = sG + (long)cur * GDIM + lane * 4;
            atomicAdd(dst + 0, acc.x);
            atomicAdd(dst + 1, acc.y);
            atomicAdd(dst + 2, acc.z);
            atomicAdd(dst + 3, acc.w);
            if (lane == 0) atomicAdd(cG + cur, gsum);
            acc = make_float4(0.f, 0.f, 0.f, 0.f);
            gsum = 0.f;
            cur = gid;
        }
        acc.x += g * hv.x; acc.y += g * hv.y;
        acc.z += g * hv.z; acc.w += g * hv.w;
        gsum  += g;
    }
    float* dst = sG + (long)cur * GDIM + lane * 4;
    atomicAdd(dst + 0, acc.x);
    atomicAdd(dst + 1, acc.y);
    atomicAdd(dst + 2, acc.z);
    atomicAdd(dst + 3, acc.w);
    if (lane == 0) atomicAdd(cG + cur, gsum);
}

// ---------------------------------------------------------------------------
// Kernel 3: out[16384,128] = sG @ W_f + outer(cG, b_f), full f32 via
// V_WMMA_F32_16X16X4_F32. Block = 256 threads = 8 waves; block owns a
// 16-row stripe, wave w owns the 16x16 tile at columns [16w, 16w+16).
//
// f32 WMMA VGPR layouts (ISA 7.12.2):
//   A 16x4 : v2f; lanes 0-15 -> row=lane, K={0,1}; lanes 16-31 -> K={2,3}
//   B 4x16 : v2f; lanes 0-15 -> col=lane, K={0,1}; lanes 16-31 -> K={2,3}
//   C/D    : v8f; VGPR r -> row r (lanes 0-15) / row r+8 (lanes 16-31)
// ---------------------------------------------------------------------------
__global__ __launch_bounds__(256) void wmma_out_kernel(
    const float* __restrict__ sG, const float* __restrict__ cG,
    const float* __restrict__ Wf, const float* __restrict__ bf,
    float* __restrict__ out)
{
    const int lane    = threadIdx.x & 31;
    const int wave    = threadIdx.x >> 5;        // 0..7 -> column tile
    const int rowBase = blockIdx.x * 16;         // M tile
    const int colBase = wave * 16;               // N tile
    const int mn      = lane & 15;               // row for A, col for B
    const int khalf   = (lane >> 4) * 2;         // 0 for lanes 0-15, 2 for 16-31

    v8f acc = {};                                // C starts at zero

    const float* aRow = sG + (long)(rowBase + mn) * NDIM;
    #pragma unroll
    for (int kk = 0; kk < NDIM / 4; ++kk) {
        const int k0 = kk * 4 + khalf;
        v2f a = *(const v2f*)(aRow + k0);        // A[m][k0], A[m][k0+1]
        v2f b;
        b.x = Wf[(long)(k0    ) * GDIM + colBase + mn];
        b.y = Wf[(long)(k0 + 1) * GDIM + colBase + mn];
        acc = __builtin_amdgcn_wmma_f32_16x16x4_f32(
                  false, a, false, b, (short)0, acc, false, false);
    }

    // fuse bias: out[row][n] = acc + cG[row] * b_f[n]
    const float bfn = bf[colBase + mn];
    #pragma unroll
    for (int r = 0; r < 8; ++r) {
        const int mr  = (lane < 16) ? r : (r + 8);
        const int row = rowBase + mr;
        out[(long)row * GDIM + colBase + mn] = acc[r] + cG[row] * bfn;
    }
}

// ---------------------------------------------------------------------------
extern "C" void kernel_launch(void* const* d_in, const int* in_sizes, int n_in,
                              void* d_out, int out_size, void* d_ws, size_t ws_size,
                              hipStream_t stream) {
    const float* h     = (const float*)d_in[0];
    const int*   batch = (const int*)  d_in[1];
    const float* W_f   = (const float*)d_in[2];
    const float* b_f   = (const float*)d_in[3];
    const float* W_g   = (const float*)d_in[4];
    const float* b_g   = (const float*)d_in[5];
    float*       out   = (float*)d_out;

    const int nNodes = in_sizes[0] / NDIM;       // 1,000,000

    // workspace layout: sG [NGRAPH*GDIM] then cG [NGRAPH]
    float* sG = (float*)d_ws;
    float* cG = sG + (size_t)NGRAPH * GDIM;
    const int wsElems = NGRAPH * GDIM + NGRAPH;

    zero_ws_kernel<<<2048, 256, 0, stream>>>(sG, wsElems);

    const int totalWaves   = 16384;              // ~61 nodes per wave
    const int nodesPerWave = (nNodes + totalWaves - 1) / totalWaves;
    gate_scatter_kernel<<<totalWaves * 32 / 256, 256, 0, stream>>>(
        h, batch, W_g, b_g, sG, cG, nNodes, nodesPerWave);

    wmma_out_kernel<<<NGRAPH / 16, 256, 0, stream>>>(sG, cG, W_f, b_f, out);
}